// GraphConv_68917045231841
// MI455X (gfx1250) — compile-verified
//
#include <hip/hip_runtime.h>

typedef __attribute__((ext_vector_type(2))) float v2f;
typedef __attribute__((ext_vector_type(8))) float v8f;

#define IN_F 96
#define OUT_F 96
// LDS W layout: K-pair interleaved, element [kk][n] = {W[2kk][n], W[2kk+1][n]} (v2f).
// stride (in v2f) chosen so 2*stride words % 64 == 32: half-wave at row kk+1 hits
// the other 32 banks -> conflict-free single ds_load_b64 per B operand.
#define LDS_STRIDE2 112     // 48 * 112 * 8B = 43008 B

// ---------------------------------------------------------------- zero ws
__global__ void zero_ws_kernel(int* __restrict__ cnt, float* __restrict__ agg,
                               int n_cnt, int n_agg) {
    int i = blockIdx.x * blockDim.x + threadIdx.x;
    if (i < n_cnt) cnt[i] = 0;
    if (i < n_agg) agg[i] = 0.0f;
}

// ---------------------------------------------------------------- degrees
__global__ void degree_kernel(const int* __restrict__ src, const int* __restrict__ dst,
                              int* __restrict__ cnt_src, int* __restrict__ cnt_dst,
                              int E) {
    int e = blockIdx.x * blockDim.x + threadIdx.x;
    if (e < E) {
        atomicAdd(&cnt_src[src[e]], 1);
        atomicAdd(&cnt_dst[dst[e]], 1);
    }
}

// ------------------------------------------------------- deg -> 1/sqrt(deg)
__global__ void invdeg_kernel(const int* __restrict__ cnt_src, const int* __restrict__ cnt_dst,
                              float* __restrict__ inv_src, float* __restrict__ inv_dst,
                              int N) {
    int i = blockIdx.x * blockDim.x + threadIdx.x;
    if (i < N) {
        int cs = cnt_src[i]; if (cs < 1) cs = 1;
        int cd = cnt_dst[i]; if (cd < 1) cd = 1;
        inv_src[i] = 1.0f / sqrtf((float)cs);
        inv_dst[i] = 1.0f / sqrtf((float)cd);
    }
}

// ------------------------------------------- edge scatter: agg[dst] += feat[src]*s
// 24 float4 chunks per edge; 24 consecutive threads cover one edge row
// (coalesced 384B gather). feat/agg are L2-resident (19.2MB each << 192MB L2),
// so the 76.8M f32 atomics are L2 RMW, not HBM traffic.
__global__ void scatter_kernel(const float4* __restrict__ feat4,
                               const int* __restrict__ src, const int* __restrict__ dst,
                               const float* __restrict__ inv_src,
                               float* __restrict__ agg, int E) {
    int idx = blockIdx.x * blockDim.x + threadIdx.x;
    int total = E * 24;
    if (idx >= total) return;
    int e = idx / 24;
    int c = idx - e * 24;
    int s = src[e];
    int d = dst[e];
    float sc = inv_src[s];
    float4 v = feat4[(size_t)s * 24 + c];
    float* o = agg + (size_t)d * IN_F + c * 4;
    atomicAdd(o + 0, v.x * sc);
    atomicAdd(o + 1, v.y * sc);
    atomicAdd(o + 2, v.z * sc);
    atomicAdd(o + 3, v.w * sc);
}

// ------------------------------------------- out = (agg @ W) * inv_dst + bias
// Full-precision f32 WMMA (V_WMMA_F32_16X16X4_F32), one wave per 16-row strip,
// 6 column tiles x 24 K-steps = 144 WMMA per wave. W staged in LDS with
// K-pair interleaving so each B operand is one ds_load_b64.
__global__ void __launch_bounds__(256)
gemm_wmma_kernel(const float* __restrict__ agg, const float* __restrict__ W,
                 const float* __restrict__ bias, const float* __restrict__ inv_dst,
                 float* __restrict__ out, int N, int n_mtiles) {
    __shared__ v2f ldsW[(IN_F / 2) * LDS_STRIDE2];

    int tid = threadIdx.x;
    // cooperative, pair-interleaved stage of W: ldsW[kk][n] = {W[2kk][n], W[2kk+1][n]}
    for (int t = tid; t < (IN_F / 2) * OUT_F; t += 256) {
        int kk = t / OUT_F;
        int n  = t - kk * OUT_F;
        v2f w;
        w.x = W[(2 * kk)     * OUT_F + n];
        w.y = W[(2 * kk + 1) * OUT_F + n];
        ldsW[kk * LDS_STRIDE2 + n] = w;
    }
    __syncthreads();

    int wave  = tid >> 5;
    int lane  = tid & 31;
    int mtile = blockIdx.x * 8 + wave;          // wave-uniform branch: EXEC stays full
    if (mtile >= n_mtiles) return;

    int row0 = mtile * 16;
    int m    = lane & 15;
    int half = lane >> 4;

    v8f acc[6] = {};

    // A row this lane feeds (affects only output row m of the tile; clamp is safe
    // because out-of-range output rows are masked at the store).
    int ar = row0 + m;
    if (ar >= N) ar = N - 1;
    const float* arow = agg + (size_t)ar * IN_F;
    int koff = 2 * half;

#pragma unroll
    for (int k0 = 0; k0 < IN_F; k0 += 4) {
        // A 16x4 f32 layout: lane m, VGPR0/1 = K = k0+2*half, k0+2*half+1
        v2f a = *(const v2f*)(arow + k0 + koff);           // 8B-aligned b64 load
        const v2f* brow = &ldsW[(k0 / 2 + half) * LDS_STRIDE2];
#pragma unroll
        for (int j = 0; j < 6; ++j) {
            // B 4x16 layout: VGPRv holds rows v and v+2 -> one v2f LDS element
            v2f b = brow[j * 16 + m];                      // single ds_load_b64
            acc[j] = __builtin_amdgcn_wmma_f32_16x16x4_f32(
                false, a, false, b, (short)0, acc[j], false, false);
        }
    }

    // Epilogue: right-normalize + bias. C/D layout: VGPRv = rows v and v+8.
    float bv[6];
#pragma unroll
    for (int j = 0; j < 6; ++j) bv[j] = bias[j * 16 + m];

#pragma unroll
    for (int v = 0; v < 8; ++v) {
        int r = row0 + v + 8 * half;
        if (r < N) {
            float s = inv_dst[r];
#pragma unroll
            for (int j = 0; j < 6; ++j)
                out[(size_t)r * OUT_F + j * 16 + m] = acc[j][v] * s + bv[j];
        }
    }
}

// ---------------------------------------------------------------- launcher
extern "C" void kernel_launch(void* const* d_in, const int* in_sizes, int n_in,
                              void* d_out, int out_size, void* d_ws, size_t ws_size,
                              hipStream_t stream) {
    (void)n_in; (void)out_size; (void)ws_size;

    const float* feat   = (const float*)d_in[0];
    const float* weight = (const float*)d_in[1];
    const float* bias   = (const float*)d_in[2];
    const int*   src    = (const int*)d_in[3];
    const int*   dst    = (const int*)d_in[4];
    float*       out    = (float*)d_out;

    int N = in_sizes[0] / IN_F;      // 50000
    int E = in_sizes[3];             // 800000

    char* ws = (char*)d_ws;
    int*   cnt_src = (int*)ws;                               // N ints
    int*   cnt_dst = (int*)(ws + (size_t)N * 4);             // N ints
    float* inv_src = (float*)(ws + (size_t)N * 8);           // N floats
    float* inv_dst = (float*)(ws + (size_t)N * 12);          // N floats
    float* agg     = (float*)(ws + (size_t)N * 16);          // N*96 floats (16B aligned)

    int n_agg = N * IN_F;

    // 1) zero counters + accumulator
    zero_ws_kernel<<<(n_agg + 255) / 256, 256, 0, stream>>>(cnt_src, agg, 2 * N, n_agg);
    // 2) degrees
    degree_kernel<<<(E + 255) / 256, 256, 0, stream>>>(src, dst, cnt_src, cnt_dst, E);
    // 3) 1/sqrt(clip(deg,1))
    invdeg_kernel<<<(N + 255) / 256, 256, 0, stream>>>(cnt_src, cnt_dst, inv_src, inv_dst, N);
    // 4) edge scatter (left-normalized)
    {
        long long total = (long long)E * 24;
        scatter_kernel<<<(int)((total + 255) / 256), 256, 0, stream>>>(
            (const float4*)feat, src, dst, inv_src, agg, E);
    }
    // 5) WMMA GEMM + right-normalization + bias
    {
        int n_mtiles = (N + 15) / 16;
        int blocks   = (n_mtiles + 7) / 8;
        gemm_wmma_kernel<<<blocks, 256, 0, stream>>>(agg, weight, bias, inv_dst, out, N, n_mtiles);
    }
}